// FrequencyAnalyzer_53712861003897
// MI455X (gfx1250) — compile-verified
//
#include <hip/hip_runtime.h>
#include <hip/hip_bf16.h>

// ---------------------------------------------------------------------------
// FrequencyAnalyzer for MI455X (gfx1250, wave32).
// Pipeline:
//   K1: row FFTs with real-pair packing (2 real rows per complex 1024-pt FFT)
//   K2: column FFTs on 16-wide slabs, slab filled via
//       GLOBAL_LOAD_ASYNC_TO_LDS_B64 (ASYNCcnt), fused |.|/1024 + radial band
//       stats (ds_add_f32 / ds_max_i32 -> global atomics)
//   K3: band stats -> 32x144 features -> MLP via V_WMMA_F32_16X16X4_F32 + LN
// Workspace: [96][1024][520] float2 intermediate (~390 MB), then
// 96*17*3 float accumulators + 96*17 int max-bits.
// ---------------------------------------------------------------------------

typedef __attribute__((ext_vector_type(2))) float v2f;
typedef __attribute__((ext_vector_type(8))) float v8f;

#define NIMG   96       // B*C = 32*3
#define NB     16       // bands
#define KSEG   17       // nb+1 segments (last one discarded)
#define HH     1024
#define WW     1024
#define WR     513      // rfft width
#define PITCH  520      // padded intermediate pitch (float2 elements)
#define COLPAD 1025     // padded LDS column stride (float2) -> no bank conflicts
#define LRELU  0.2f

// 1024-point radix-2 Stockham FFT (DIF, autosorting). 256 threads cooperate,
// ping-pong between buf and scr; 10 stages (even) -> result lands back in buf
// in natural order. Forward transform (w = exp(-2*pi*i*p/n)).
__device__ __forceinline__ void fft1024(float2* __restrict__ buf,
                                        float2* __restrict__ scr, int tid) {
  float2* src = buf;
  float2* dst = scr;
  int n = 1024, s = 1, ls = 0;
  while (n > 1) {
    const int m = n >> 1;
#pragma unroll
    for (int jj = 0; jj < 2; ++jj) {
      const int j = tid + jj * 256;          // 512 butterflies / stage
      const int p = j >> ls;                 // [0, m)
      const int q = j & (s - 1);             // [0, s)
      const float ang = -6.28318530717958647692f * (float)p / (float)n;
      float sn, cs;
      __sincosf(ang, &sn, &cs);
      const float2 u = src[q + s * p];
      const float2 w = src[q + s * p + s * m];
      const float dx = u.x - w.x, dy = u.y - w.y;
      dst[q + 2 * s * p]     = make_float2(u.x + w.x, u.y + w.y);
      dst[q + 2 * s * p + s] = make_float2(dx * cs - dy * sn, dx * sn + dy * cs);
    }
    __syncthreads();
    float2* t = src; src = dst; dst = t;
    n = m; s <<= 1; ++ls;
  }
}

// --------------------------- kernel 0: reset -------------------------------
__global__ void freq_zero_acc(float* __restrict__ gacc, int* __restrict__ gmax) {
  const int i = blockIdx.x * 256 + threadIdx.x;
  if (i < NIMG * KSEG * 3) gacc[i] = 0.0f;
  if (i < NIMG * KSEG)     gmax[i] = 0;      // bits of +0.0f
}

// ------------- kernel 1: row FFTs with real-pair packing -------------------
// grid = (512 row pairs, 96 images), 256 threads.
// z = rowA + i*rowB; one complex FFT yields both real-row spectra:
//   FA[k] = 0.5*(Z[k] + conj(Z[N-k]))
//   FB[k] = (Z[k] - conj(Z[N-k])) / (2i)
__global__ void freq_rowfft(const float* __restrict__ x,
                            float2* __restrict__ interm) {
  __shared__ float2 A[1024];
  __shared__ float2 Bb[1024];
  const int pr  = blockIdx.x;          // row pair
  const int img = blockIdx.y;
  const int tid = threadIdx.x;
  const int r0 = 2 * pr, r1 = r0 + 1;
  const float* xr0 = x + ((size_t)img * HH + r0) * WW;
  const float* xr1 = x + ((size_t)img * HH + r1) * WW;
#pragma unroll
  for (int c = tid; c < 1024; c += 256)
    A[c] = make_float2(xr0[c], xr1[c]);
  __syncthreads();
  fft1024(A, Bb, tid);
  float2* o0 = interm + ((size_t)img * HH + r0) * PITCH;
  float2* o1 = interm + ((size_t)img * HH + r1) * PITCH;
  for (int k = tid; k < WR; k += 256) {
    const int nk = (1024 - k) & 1023;
    const float2 zk = A[k];
    const float2 zn = A[nk];
    o0[k] = make_float2(0.5f * (zk.x + zn.x), 0.5f * (zk.y - zn.y));
    o1[k] = make_float2(0.5f * (zk.y + zn.y), 0.5f * (zn.x - zk.x));
  }
}

// ---------- kernel 2: column FFTs + fused magnitude band statistics --------
// grid = (33 column tiles of 16, 96 images), 256 threads,
// dynamic LDS = (16*1025 + 1024) float2 = ~136 KB (2 blocks per 320KB WGP).
// Slab fill uses the CDNA5 async memory->LDS path (ASYNCcnt).
__global__ void freq_colfft_stats(const float2* __restrict__ interm,
                                  float* __restrict__ gacc,
                                  int* __restrict__ gmax) {
  extern __shared__ unsigned char s_dyn[];
  float2* slab = (float2*)s_dyn;                 // [16][1025]
  float2* scr  = slab + 16 * COLPAD;             // [1024]
  __shared__ float lcnt[KSEG], ls1[KSEG], ls2[KSEG];
  __shared__ int lmx[KSEG];

  const int tid = threadIdx.x;
  const int img = blockIdx.y;
  const int c0  = blockIdx.x * 16;
  const int nc  = min(16, WR - c0);
  if (tid < KSEG) { lcnt[tid] = 0.f; ls1[tid] = 0.f; ls2[tid] = 0.f; lmx[tid] = 0; }

  const float2* base = interm + (size_t)img * HH * PITCH;
  const int total = 1024 * nc;                   // multiple of 256: EXEC full
  for (int idx = tid; idx < total; idx += 256) {
    const int rowq = idx / nc;
    const int c    = idx - rowq * nc;
    const unsigned long long gaddr =
        (unsigned long long)(const void*)&base[(size_t)rowq * PITCH + c0 + c];
    const unsigned ldsoff = (unsigned)(size_t)(void*)&slab[c * COLPAD + rowq];
    // async global -> LDS copy, tracked on ASYNCcnt (no VGPR staging)
    asm volatile("global_load_async_to_lds_b64 %0, %1, off"
                 :: "v"(ldsoff), "v"(gaddr) : "memory");
  }
  asm volatile("s_wait_asynccnt 0x0" ::: "memory");
  __syncthreads();

  const float inv_bw = (float)NB / 0.70710678118654752440f;  // nb / max_r
  for (int c = 0; c < nc; ++c) {
    float2* col = slab + c * COLPAD;
    fft1024(col, scr, tid);
    const float u = (float)(c0 + c) * (1.0f / (float)WW);    // rfftfreq
    for (int r = tid; r < 1024; r += 256) {
      const float2 vv = col[r];
      const float mag = sqrtf(vv.x * vv.x + vv.y * vv.y) * (1.0f / 1024.0f); // ortho
      const float vf  = ((r < 512) ? (float)r : (float)(r - 1024)) * (1.0f / (float)HH);
      const float rad = sqrtf(u * u + vf * vf);
      int bin = (int)(rad * inv_bw);
      bin = (bin > NB) ? NB : bin;
      atomicAdd(&lcnt[bin], 1.0f);         // ds_add_f32
      atomicAdd(&ls1[bin], mag);
      atomicAdd(&ls2[bin], mag * mag);
      atomicMax(&lmx[bin], __float_as_int(mag));   // mags >= 0: int order == float order
    }
  }
  __syncthreads();
  if (tid < KSEG) {
    float* a = gacc + (img * KSEG + tid) * 3;
    atomicAdd(&a[0], lcnt[tid]);
    atomicAdd(&a[1], ls1[tid]);
    atomicAdd(&a[2], ls2[tid]);
    atomicMax(&gmax[img * KSEG + tid], lmx[tid]);
  }
}

// ------------- kernel 3: band stats -> features -> MLP (WMMA f32) ----------
// single block, 256 threads (8 wave32).
// GEMM tiling: M=32 (2 tiles), N=256 (16 tiles), K steps of 4 via
// V_WMMA_F32_16X16X4_F32. Fragment layouts per CDNA5 ISA 7.12.2:
//   A (16x4 f32): lanes 0-15 hold (M=lane, K=0/1 in v0/v1),
//                 lanes 16-31 hold (M=lane-16, K=2/3 in v0/v1).
//   B (4x16 f32): mirrored (rows K striped the same way, N = lane&15).
//   C/D (16x16):  vgpr r, lanes 0-15 -> (M=r, N=lane); lanes 16-31 -> (M=r+8).
__global__ void freq_mlp_wmma(const float* __restrict__ gacc,
                              const int* __restrict__ gmax,
                              const float* __restrict__ W1,
                              const float* __restrict__ b1,
                              const float* __restrict__ W2,
                              const float* __restrict__ b2,
                              const float* __restrict__ gamma,
                              const float* __restrict__ beta,
                              float* __restrict__ out) {
  extern __shared__ unsigned char s_dyn[];
  float* F  = (float*)s_dyn;        // [32][148] features, fin=144
  float* h1 = F  + 32 * 148;        // [32][256]
  float* h2 = h1 + 32 * 256;        // [32][256]
  float* mu = h2 + 32 * 256;        // [32]
  float* rs = mu + 32;              // [32]
  const int tid = threadIdx.x;

  // features: f = (k*3 + stat)*3 + c, stat in {mean, max, std}
  for (int idx = tid; idx < 32 * 144; idx += 256) {
    const int b = idx / 144, f = idx - b * 144;
    const int k = f / 9, rem = f - k * 9;
    const int s = rem / 3, c = rem - s * 3;
    const int img = b * 3 + c;
    const float* a = gacc + (img * KSEG + k) * 3;
    const float cnt = a[0], s1 = a[1], s2 = a[2];
    const float denom = cnt + 1e-8f;
    const float mean = s1 / denom;
    const float var  = (s2 - 2.0f * mean * s1 + cnt * mean * mean) / denom;
    float val;
    if (s == 0)      val = mean;
    else if (s == 1) val = fmaxf(__int_as_float(gmax[img * KSEG + k]), 0.0f);
    else             val = sqrtf(fmaxf(var, 0.0f));
    F[b * 148 + f] = val;
  }
  __syncthreads();

  const int lane  = tid & 31;
  const int wave  = tid >> 5;
  const int koffs = (lane < 16) ? 0 : 2;
  const int mrow  = lane & 15;

  // layer 1: h1 = lrelu(F(32x144) @ W1(144x256) + b1)
  for (int t = wave; t < 32; t += 8) {           // uniform per wave
    const int mt = t & 1, nt = t >> 1;
    v8f acc = {};
    for (int k0 = 0; k0 < 144; k0 += 4) {
      const float* fr = &F[(mt * 16 + mrow) * 148 + k0 + koffs];
      v2f av = { fr[0], fr[1] };
      const float* wr = &W1[(size_t)(k0 + koffs) * 256 + nt * 16 + mrow];
      v2f bv = { wr[0], wr[256] };
      acc = __builtin_amdgcn_wmma_f32_16x16x4_f32(false, av, false, bv,
                                                  (short)0, acc, false, false);
    }
    const int mbase = mt * 16 + ((lane < 16) ? 0 : 8);
    const int n = nt * 16 + mrow;
    const float bb = b1[n];
#pragma unroll
    for (int r = 0; r < 8; ++r) {
      float v = acc[r] + bb;
      v = (v >= 0.0f) ? v : LRELU * v;
      h1[(mbase + r) * 256 + n] = v;
    }
  }
  __syncthreads();

  // layer 2: h2 = h1(32x256) @ W2(256x256) + b2
  for (int t = wave; t < 32; t += 8) {
    const int mt = t & 1, nt = t >> 1;
    v8f acc = {};
    for (int k0 = 0; k0 < 256; k0 += 4) {
      const float* fr = &h1[(mt * 16 + mrow) * 256 + k0 + koffs];
      v2f av = { fr[0], fr[1] };
      const float* wr = &W2[(size_t)(k0 + koffs) * 256 + nt * 16 + mrow];
      v2f bv = { wr[0], wr[256] };
      acc = __builtin_amdgcn_wmma_f32_16x16x4_f32(false, av, false, bv,
                                                  (short)0, acc, false, false);
    }
    const int mbase = mt * 16 + ((lane < 16) ? 0 : 8);
    const int n = nt * 16 + mrow;
    const float bb = b2[n];
#pragma unroll
    for (int r = 0; r < 8; ++r) h2[(mbase + r) * 256 + n] = acc[r] + bb;
  }
  __syncthreads();

  // layernorm over last dim (256) per row
  if (tid < 32) {
    const float* hr = &h2[tid * 256];
    float s = 0.0f;
    for (int j = 0; j < 256; ++j) s += hr[j];
    const float m = s * (1.0f / 256.0f);
    float v = 0.0f;
    for (int j = 0; j < 256; ++j) { const float d = hr[j] - m; v += d * d; }
    mu[tid] = m;
    rs[tid] = rsqrtf(v * (1.0f / 256.0f) + 1e-5f);
  }
  __syncthreads();
  for (int idx = tid; idx < 32 * 256; idx += 256) {
    const int b = idx >> 8, n = idx & 255;
    out[idx] = (h2[idx] - mu[b]) * rs[b] * gamma[n] + beta[n];
  }
}

// ---------------------------------------------------------------------------
extern "C" void kernel_launch(void* const* d_in, const int* in_sizes, int n_in,
                              void* d_out, int out_size, void* d_ws, size_t ws_size,
                              hipStream_t stream) {
  const float* x     = (const float*)d_in[0];
  const float* W1    = (const float*)d_in[1];
  const float* b1    = (const float*)d_in[2];
  const float* W2    = (const float*)d_in[3];
  const float* b2    = (const float*)d_in[4];
  const float* gamma = (const float*)d_in[5];
  const float* beta  = (const float*)d_in[6];
  float* out = (float*)d_out;

  char* ws = (char*)d_ws;
  float2* interm = (float2*)ws;
  const size_t intermBytes = (size_t)NIMG * HH * PITCH * sizeof(float2);
  float* gacc = (float*)(ws + intermBytes);
  int*   gmax = (int*)(ws + intermBytes + (size_t)NIMG * KSEG * 3 * sizeof(float));

  // accumulators must be re-zeroed every launch (harness does not re-poison)
  freq_zero_acc<<<(NIMG * KSEG * 3 + 255) / 256, 256, 0, stream>>>(gacc, gmax);

  freq_rowfft<<<dim3(HH / 2, NIMG), 256, 0, stream>>>(x, interm);

  const size_t ldsK2 = (size_t)(16 * COLPAD + 1024) * sizeof(float2); // ~136 KB
  freq_colfft_stats<<<dim3((WR + 15) / 16, NIMG), 256, ldsK2, stream>>>(interm, gacc, gmax);

  const size_t ldsK3 = (size_t)(32 * 148 + 32 * 256 + 32 * 256 + 64) * sizeof(float);
  freq_mlp_wmma<<<1, 256, ldsK3, stream>>>(gacc, gmax, W1, b1, W2, b2,
                                           gamma, beta, out);
}